// TaskAlignedAssigner_23519240913065
// MI455X (gfx1250) — compile-verified
//
#include <hip/hip_runtime.h>
#include <hip/hip_bf16.h>

#define BS    16
#define NA    33600
#define NMAX  64
#define NCLS  80
#define TOPKN 13
#define CAP   1024
#define EPSV  1e-9f

typedef __attribute__((ext_vector_type(2))) float v2f;
typedef __attribute__((ext_vector_type(8))) float v8f;

// D = A(16x4) * B(4x16) + C, f32. 8-arg form: (neg_a, A, neg_b, B, c_mod, C, reuse_a, reuse_b)
__device__ __forceinline__ v8f wmma_16x16x4(v2f a, v2f b, v8f c) {
    return __builtin_amdgcn_wmma_f32_16x16x4_f32(false, a, false, b, (short)0, c, false, false);
}

// 1-ulp hardware transcendentals (v_sqrt_f32 / v_rcp_f32, co-execute as TRANS ops)
__device__ __forceinline__ float fsqrt(float x) { return __builtin_amdgcn_sqrtf(x); }
__device__ __forceinline__ float frcp(float x)  { return __builtin_amdgcn_rcpf(x); }

// ---------------------------------------------------------------- zero scratch
__global__ void k_zero(int* __restrict__ acnt, unsigned* __restrict__ pos_al,
                       unsigned* __restrict__ pos_ov) {
    size_t i = (size_t)blockIdx.x * blockDim.x + threadIdx.x;
    if (i < (size_t)BS * NA) acnt[i] = 0;
    if (i < (size_t)BS * NMAX) { pos_al[i] = 0u; pos_ov[i] = 0u; }
}

// ---------------------------------------------------------------- fused metric + top-k
// grid = (NMAX/16, BS), 256 threads (8 waves). Block handles 16 gts over all anchors.
__global__ void __launch_bounds__(256) k_topk(
    const float* __restrict__ pd_scores, const float* __restrict__ pd_circles,
    const float* __restrict__ anc, const int* __restrict__ gt_labels,
    const float* __restrict__ gt_bboxes, const float* __restrict__ mask_gt,
    float* __restrict__ sel_metric, int* __restrict__ sel_idx,
    float* __restrict__ sel_over, int* __restrict__ sel_cnt)
{
    __shared__ unsigned long long s_list[16 * CAP];   // 128 KB candidate keys (WGP has 320 KB)
    __shared__ unsigned long long s_win[16 * TOPKN];
    __shared__ int   s_cnt[16], s_nwin[16], s_lab[16];
    __shared__ float s_gx[16], s_gy[16], s_gr[16], s_gr2m[16];  // gr2m: gr^2, or -1 if masked off

    const int b     = blockIdx.y;
    const int gbase = blockIdx.x * 16;
    const int tid   = threadIdx.x;
    const int lane  = tid & 31;
    const int wave  = tid >> 5;
    const int half  = lane >> 4;   // K-half for A/B fragments
    const int nm    = lane & 15;   // row (A) / column (B,C,D)

    if (tid < 16) {
        int g = gbase + tid;
        const float* gb = gt_bboxes + ((size_t)b * NMAX + g) * 3;
        float gx = gb[0], gy = gb[1], gr = gb[2];
        float mk = mask_gt[(size_t)b * NMAX + g];
        s_gx[tid] = gx; s_gy[tid] = gy; s_gr[tid] = gr;
        s_gr2m[tid] = (mk > 0.0f) ? gr * gr : -1.0f;   // fold mask_gt into radius test
        s_lab[tid] = gt_labels[(size_t)b * NMAX + g];
        s_cnt[tid] = 0;
    }
    __syncthreads();

    // A fragment (constant over anchor loop): row m = [gx, gy, gx^2+gy^2, 1]
    float gx = s_gx[nm], gy = s_gy[nm];
    v2f afrag;
    afrag.x = half ? (gx * gx + gy * gy) : gx;
    afrag.y = half ? 1.0f : gy;

    // each lane only ever touches rows m = half*8 .. half*8+7 -> keep them in registers
    float r_gr2[8], r_gr[8];
    int   r_lab[8];
#pragma unroll
    for (int v = 0; v < 8; ++v) {
        int m = v + half * 8;
        r_gr2[v] = s_gr2m[m]; r_gr[v] = s_gr[m]; r_lab[v] = s_lab[m];
    }
    const float* scb = pd_scores + (size_t)b * NA * NCLS;

    const int ntiles = NA / 16;  // 2100
    for (int t = wave; t < ntiles; t += 8) {
        int a0 = t * 16 + nm;
        float ax = anc[(size_t)a0 * 2 + 0];
        float ay = anc[(size_t)a0 * 2 + 1];
        const float* cc = pd_circles + ((size_t)b * NA + a0) * 3;
        float cx = cc[0], cy = cc[1], cr = cc[2];
        __builtin_prefetch(pd_circles + ((size_t)b * NA + (a0 + 128 < NA ? a0 + 128 : a0)) * 3, 0, 1);

        // B col n = [-2px, -2py, 1, px^2+py^2]  ->  D = squared distance
        v2f banc, bctr;
        banc.x = half ? 1.0f : (-2.0f * ax);
        banc.y = half ? (ax * ax + ay * ay) : (-2.0f * ay);
        bctr.x = half ? 1.0f : (-2.0f * cx);
        bctr.y = half ? (cx * cx + cy * cy) : (-2.0f * cy);

        v8f zero = 0;
        v8f d2anc = wmma_16x16x4(afrag, banc, zero);  // gt <-> anchor point
        v8f d2ctr = wmma_16x16x4(afrag, bctr, zero);  // gt <-> pred center

#pragma unroll
        for (int v = 0; v < 8; ++v) {
            int m = v + half * 8;                      // C/D layout: M = vgpr + 8*(lane/16)
            if (d2anc[v] <= r_gr2[v]) {                // mask_gt pre-folded (-1 fails always)
                float d   = fsqrt(fmaxf(d2ctr[v], 0.0f));
                float rs  = r_gr[v] + cr;
                float iou = fmaxf((rs - d) * frcp(rs), 0.0f);
                float sc  = scb[(size_t)a0 * NCLS + r_lab[v]];
                float i3  = iou * iou * iou;
                float met = sc * i3 * i3;              // ALPHA=1, BETA=6
                int pos = atomicAdd(&s_cnt[m], 1);
                if (pos < CAP)
                    s_list[m * CAP + pos] =
                        ((unsigned long long)__float_as_uint(met) << 32) | (unsigned)(~a0);
            }
        }
    }
    __syncthreads();

    // top-13 selection: wave w handles rows 2w, 2w+1 via shuffle max-reduction
    for (int rr = 0; rr < 2; ++rr) {
        int m  = wave * 2 + rr;
        int nc = min(s_cnt[m], CAP);
        int nw = 0;
        for (int k = 0; k < TOPKN; ++k) {
            unsigned long long lmax = 0ull; int lslot = -1;
            for (int s = lane; s < nc; s += 32) {
                unsigned long long v = s_list[m * CAP + s];
                if (v > lmax) { lmax = v; lslot = s; }
            }
            unsigned long long gmax = lmax;
            for (int off = 16; off > 0; off >>= 1) {
                unsigned long long o = __shfl_xor(gmax, off, 32);
                if (o > gmax) gmax = o;
            }
            if (gmax == 0ull) break;                 // no candidates left
            if (lmax == gmax && lslot >= 0) s_list[m * CAP + lslot] = 0ull;  // unique key
            if (lane == 0) s_win[m * TOPKN + k] = gmax;
            nw++;
        }
        if (lane == 0) {
            float m0 = (nw > 0) ? __uint_as_float((unsigned)(s_win[m * TOPKN] >> 32)) : 0.0f;
            s_nwin[m] = (m0 > EPSV) ? nw : 0;        // topk_mask: row max > EPS
        }
    }
    __syncthreads();

    // emit winners; recompute overlap only for the <=13 selected anchors
    for (int t = tid; t < 16 * TOPKN; t += 256) {
        int m = t / TOPKN, j = t % TOPKN;
        int bg = b * NMAX + gbase + m;
        int cnt = s_nwin[m];
        if (j == 0) sel_cnt[bg] = cnt;
        if (j < cnt) {
            unsigned long long key = s_win[m * TOPKN + j];
            int a = (int)(~(unsigned)(key & 0xFFFFFFFFull));
            float met = __uint_as_float((unsigned)(key >> 32));
            const float* cc = pd_circles + ((size_t)b * NA + a) * 3;
            float dx = s_gx[m] - cc[0], dy = s_gy[m] - cc[1];
            float d  = fsqrt(dx * dx + dy * dy);
            float rs = s_gr[m] + cc[2];
            float ov = fmaxf((rs - d) * frcp(rs), 0.0f);
            sel_idx[bg * 16 + j]    = a;
            sel_metric[bg * 16 + j] = met;
            sel_over[bg * 16 + j]   = ov;
        }
    }
}

// ---------------------------------------------------------------- scatter selections to anchors
__global__ void k_scatter(const int* __restrict__ sel_cnt, const int* __restrict__ sel_idx,
                          int* __restrict__ acnt, int* __restrict__ agt) {
    int w = blockIdx.x * blockDim.x + threadIdx.x;
    if (w >= BS * NMAX) return;
    int b = w / NMAX, g = w % NMAX;
    int c = sel_cnt[w];
    for (int j = 0; j < c; ++j) {
        int a = sel_idx[w * 16 + j];
        atomicAdd(&acnt[(size_t)b * NA + a], 1);
        agt[(size_t)b * NA + a] = g;   // unique writer when final count == 1
    }
}

// ---------------------------------------------------------------- dedup + per-gt norm maxima
__global__ void __launch_bounds__(256) k_resolve(
    const float* __restrict__ pd_scores, const float* __restrict__ pd_circles,
    const float* __restrict__ anc, const int* __restrict__ gt_labels,
    const float* __restrict__ gt_bboxes, const float* __restrict__ mask_gt,
    const int* __restrict__ acnt, int* __restrict__ agt, int* __restrict__ afg,
    float* __restrict__ ametric, unsigned* __restrict__ pos_al, unsigned* __restrict__ pos_ov)
{
    __shared__ float s_gx[NMAX], s_gy[NMAX], s_gr[NMAX], s_gr2m[NMAX];
    __shared__ int   s_lab[NMAX];
    int b = blockIdx.y, tid = threadIdx.x;
    if (tid < NMAX) {
        const float* gb = gt_bboxes + ((size_t)b * NMAX + tid) * 3;
        float gx = gb[0], gy = gb[1], gr = gb[2];
        float mk = mask_gt[(size_t)b * NMAX + tid];
        s_gx[tid] = gx; s_gy[tid] = gy; s_gr[tid] = gr;
        s_gr2m[tid] = (mk > 0.0f) ? gr * gr : -1.0f;
        s_lab[tid] = gt_labels[(size_t)b * NMAX + tid];
    }
    __syncthreads();
    int a = blockIdx.x * 256 + tid;
    if (a >= NA) return;
    size_t idx = (size_t)b * NA + a;
    int c = acnt[idx];
    int g = 0, fg = 0; float met = 0.0f;
    if (c > 0) {
        float ax = anc[(size_t)a * 2], ay = anc[(size_t)a * 2 + 1];
        const float* cc = pd_circles + idx * 3;
        float cx = cc[0], cy = cc[1], cr = cc[2];
        if (c == 1) { g = agt[idx]; }
        else {  // multi-assigned: argmax over masked overlaps across ALL gts (first-max ties)
            float best = -1.0f;
            for (int gg = 0; gg < NMAX; ++gg) {
                float dxa = s_gx[gg] - ax, dya = s_gy[gg] - ay;
                float ov = 0.0f;
                if (dxa * dxa + dya * dya <= s_gr2m[gg]) {
                    float dx = s_gx[gg] - cx, dy = s_gy[gg] - cy;
                    float d  = fsqrt(dx * dx + dy * dy);
                    float rs = s_gr[gg] + cr;
                    ov = fmaxf((rs - d) * frcp(rs), 0.0f);
                }
                if (ov > best) { best = ov; g = gg; }
            }
        }
        fg = 1;
        float dxa = s_gx[g] - ax, dya = s_gy[g] - ay;
        float ov = 0.0f;
        if (dxa * dxa + dya * dya <= s_gr2m[g]) {
            float dx = s_gx[g] - cx, dy = s_gy[g] - cy;
            float d  = fsqrt(dx * dx + dy * dy);
            float rs = s_gr[g] + cr;
            float iou = fmaxf((rs - d) * frcp(rs), 0.0f);
            float sc  = pd_scores[idx * NCLS + s_lab[g]];
            float i3  = iou * iou * iou;
            met = sc * i3 * i3;
            ov  = iou;
        }
        atomicMax(&pos_al[b * NMAX + g], __float_as_uint(met)); // monotone for floats >= 0
        atomicMax(&pos_ov[b * NMAX + g], __float_as_uint(ov));
    }
    agt[idx] = g; afg[idx] = fg; ametric[idx] = met;
}

// ---------------------------------------------------------------- final outputs
__global__ void __launch_bounds__(256) k_out(
    const int* __restrict__ gt_labels, const float* __restrict__ gt_bboxes,
    const int* __restrict__ agt, const int* __restrict__ afg,
    const float* __restrict__ ametric, const unsigned* __restrict__ pos_al,
    const unsigned* __restrict__ pos_ov, float* __restrict__ out)
{
    int b = blockIdx.y;
    int a = blockIdx.x * 256 + threadIdx.x;
    if (a >= NA) return;
    size_t idx  = (size_t)b * NA + a;
    const size_t NT = (size_t)BS * NA;
    int g  = agt[idx];
    int fg = afg[idx];
    int lab = max(gt_labels[(size_t)b * NMAX + g], 0);
    const float* gb = gt_bboxes + ((size_t)b * NMAX + g) * 3;
    out[idx] = (float)lab;                               // target_labels
    float* obox = out + NT + idx * 3;                    // target_bboxes
    obox[0] = gb[0]; obox[1] = gb[1]; obox[2] = gb[2];
    float norm = 0.0f;
    if (fg) {
        float pa = __uint_as_float(pos_al[b * NMAX + g]);
        float po = __uint_as_float(pos_ov[b * NMAX + g]);
        norm = ametric[idx] * po * frcp(pa + EPSV);
    }
    float* osc = out + NT * 4 + idx * NCLS;              // target_scores
    for (int i = 0; i < NCLS; ++i) osc[i] = 0.0f;
    if (fg) osc[lab] = norm;
    out[NT * 84 + idx] = fg ? 1.0f : 0.0f;               // fg_mask
    out[NT * 85 + idx] = (float)g;                       // target_gt_idx
}

// ---------------------------------------------------------------- host
extern "C" void kernel_launch(void* const* d_in, const int* in_sizes, int n_in,
                              void* d_out, int out_size, void* d_ws, size_t ws_size,
                              hipStream_t stream) {
    (void)in_sizes; (void)n_in; (void)out_size; (void)ws_size;
    const float* pd_scores  = (const float*)d_in[0];
    const float* pd_circles = (const float*)d_in[1];
    const float* anc        = (const float*)d_in[2];
    const int*   gt_labels  = (const int*)d_in[3];
    const float* gt_bboxes  = (const float*)d_in[4];
    const float* mask_gt    = (const float*)d_in[5];
    float* out = (float*)d_out;

    char* p = (char*)d_ws;
    auto carve = [&](size_t bytes) { void* r = (void*)p; p += (bytes + 255) & ~(size_t)255; return r; };
    float*    sel_metric = (float*)carve((size_t)BS * NMAX * 16 * sizeof(float));
    int*      sel_idx    = (int*)  carve((size_t)BS * NMAX * 16 * sizeof(int));
    float*    sel_over   = (float*)carve((size_t)BS * NMAX * 16 * sizeof(float));
    int*      sel_cnt    = (int*)  carve((size_t)BS * NMAX * sizeof(int));
    int*      acnt       = (int*)  carve((size_t)BS * NA * sizeof(int));
    int*      agt        = (int*)  carve((size_t)BS * NA * sizeof(int));
    int*      afg        = (int*)  carve((size_t)BS * NA * sizeof(int));
    float*    ametric    = (float*)carve((size_t)BS * NA * sizeof(float));
    unsigned* pos_al     = (unsigned*)carve((size_t)BS * NMAX * sizeof(unsigned));
    unsigned* pos_ov     = (unsigned*)carve((size_t)BS * NMAX * sizeof(unsigned));

    int nblk = (BS * NA + 255) / 256;
    k_zero<<<nblk, 256, 0, stream>>>(acnt, pos_al, pos_ov);
    k_topk<<<dim3(NMAX / 16, BS), 256, 0, stream>>>(pd_scores, pd_circles, anc, gt_labels,
                                                    gt_bboxes, mask_gt,
                                                    sel_metric, sel_idx, sel_over, sel_cnt);
    k_scatter<<<(BS * NMAX + 255) / 256, 256, 0, stream>>>(sel_cnt, sel_idx, acnt, agt);
    dim3 ga((NA + 255) / 256, BS);
    k_resolve<<<ga, 256, 0, stream>>>(pd_scores, pd_circles, anc, gt_labels, gt_bboxes, mask_gt,
                                      acnt, agt, afg, ametric, pos_al, pos_ov);
    k_out<<<ga, 256, 0, stream>>>(gt_labels, gt_bboxes, agt, afg, ametric, pos_al, pos_ov, out);
}